// MSFlowMatching_25864293056766
// MI455X (gfx1250) — compile-verified
//
#include <hip/hip_runtime.h>
#include <hip/hip_bf16.h>
#include <math.h>

// ---------------------------------------------------------------------------
// MSFlowMatching loss, fused for MI455X (gfx1250, wave32).
//  - Pairwise-distance Gram tiles via V_WMMA_F32_16X16X4_F32 (K=3 padded to 4)
//  - Fully fused: the 16x768x768 distance matrices are never materialized
//    (saves ~226 MB of HBM traffic; everything stays in VGPRs/L2)
// ---------------------------------------------------------------------------

constexpr int   BB   = 16;
constexpr int   LL   = 768;
constexpr int   NT   = 256;          // threads per block (8 waves @ wave32)
constexpr int   EPT  = LL / NT;      // 3 mask elements per thread in the scan
constexpr int   TILES = LL / 16;     // 48 tiles per dimension
constexpr int   WAVES_PER_BLK = 8;   // one 16x16 tile per wave
constexpr int   BLK_PER_BATCH = (TILES * TILES) / WAVES_PER_BLK; // 288
constexpr float EPSF = 1e-6f;

typedef __attribute__((ext_vector_type(2))) float v2f;
typedef __attribute__((ext_vector_type(8))) float v8f;

// sin/cos of CA angle at vertex j+1 (sin(arccos x)=sqrt(1-x^2), exact identity)
__device__ inline void angle_sc(const float* c, int j, float& sn, float& cs) {
  float ax = c[3*j+0] - c[3*j+3], ay = c[3*j+1] - c[3*j+4], az = c[3*j+2] - c[3*j+5];
  float bx = c[3*j+6] - c[3*j+3], by = c[3*j+7] - c[3*j+4], bz = c[3*j+8] - c[3*j+5];
  float d  = ax*bx + ay*by + az*bz;
  float na = sqrtf(ax*ax + ay*ay + az*az);
  float nb = sqrtf(bx*bx + by*by + bz*bz);
  float x  = d / (na*nb + EPSF);
  x  = fminf(1.f, fmaxf(-1.f, x));
  cs = x;
  sn = sqrtf(fmaxf(0.f, 1.f - x*x));
}

// sin/cos of signed dihedral at j..j+3
__device__ inline void dihedral_sc(const float* c, int j, float& sn, float& cs) {
  float v1x = c[3*j+3]-c[3*j+0], v1y = c[3*j+4]-c[3*j+1], v1z = c[3*j+5]-c[3*j+2];
  float v2x = c[3*j+6]-c[3*j+3], v2y = c[3*j+7]-c[3*j+4], v2z = c[3*j+8]-c[3*j+5];
  float v3x = c[3*j+9]-c[3*j+6], v3y = c[3*j+10]-c[3*j+7], v3z = c[3*j+11]-c[3*j+8];
  float n1x = v1y*v2z - v1z*v2y, n1y = v1z*v2x - v1x*v2z, n1z = v1x*v2y - v1y*v2x;
  float n2x = v2y*v3z - v2z*v3y, n2y = v2z*v3x - v2x*v3z, n2z = v2x*v3y - v2y*v3x;
  float d   = n1x*n2x + n1y*n2y + n1z*n2z;
  float a1  = sqrtf(n1x*n1x + n1y*n1y + n1z*n1z);
  float a2  = sqrtf(n2x*n2x + n2y*n2y + n2z*n2z);
  float x   = d / (a1*a2 + EPSF);
  x = fminf(1.f, fmaxf(-1.f, x));
  // sign = sign(dot(cross(n1,n2), v2))
  float cx = n1y*n2z - n1z*n2y, cy = n1z*n2x - n1x*n2z, cz = n1x*n2y - n1y*n2x;
  float sg = cx*v2x + cy*v2y + cz*v2z;
  float s  = (sg > 0.f) ? 1.f : ((sg < 0.f) ? -1.f : 0.f);
  if (s == 0.f) { sn = 0.f; cs = 1.f; }           // sign*phi == 0 exactly
  else          { sn = s * sqrtf(fmaxf(0.f, 1.f - x*x)); cs = x; }
}

// ---------------------------------------------------------------------------
// Kernel 1: per-batch stable compaction + point norms + angle/torsion sums.
// One block per batch. Also zero-inits the dist/clash accumulators.
// acc layout (per batch, 8 floats): [angle_num, angle_den, tors_num, tors_den,
//                                    dist_num(atomic), clash_num(atomic), -, -]
// ---------------------------------------------------------------------------
__global__ void compact_angles_kernel(const float* __restrict__ pred,
                                      const float* __restrict__ target,
                                      const int*   __restrict__ gmask,
                                      float* __restrict__ pred_e,
                                      float* __restrict__ target_e,
                                      float* __restrict__ npred,
                                      float* __restrict__ ntarg,
                                      int*   __restrict__ cnt_out,
                                      float* __restrict__ acc) {
  __shared__ float sp[LL * 3];
  __shared__ float st[LL * 3];
  __shared__ int   sscan[NT];
  __shared__ float red[NT];

  const int b = blockIdx.x, t = threadIdx.x;
  const float* pin  = pred   + (size_t)b * LL * 3;
  const float* tin  = target + (size_t)b * LL * 3;
  const int*   mrow = gmask  + (size_t)b * LL;

  // --- stable-compaction prefix scan (matches stable argsort of 0/1 keys) ---
  int m[EPT], pref[EPT], ssum = 0;
#pragma unroll
  for (int e = 0; e < EPT; ++e) {
    pref[e] = ssum;
    m[e] = (mrow[t * EPT + e] > 0) ? 1 : 0;
    ssum += m[e];
  }
  sscan[t] = ssum;
  __syncthreads();
  for (int off = 1; off < NT; off <<= 1) {
    int v = (t >= off) ? sscan[t - off] : 0;
    __syncthreads();
    sscan[t] += v;
    __syncthreads();
  }
  const int base = sscan[t] - ssum;
  const int cnt  = sscan[NT - 1];

  for (int i = t; i < LL * 3; i += NT) { sp[i] = 0.f; st[i] = 0.f; }
  __syncthreads();
#pragma unroll
  for (int e = 0; e < EPT; ++e) {
    if (m[e]) {
      int src = t * EPT + e, dst = base + pref[e];
#pragma unroll
      for (int d = 0; d < 3; ++d) {
        sp[dst * 3 + d] = pin[src * 3 + d];
        st[dst * 3 + d] = tin[src * 3 + d];
      }
    }
  }
  __syncthreads();

  // --- write compacted coords + squared norms for the cdist stage ---
  for (int i = t; i < LL * 3; i += NT) {
    pred_e[(size_t)b * LL * 3 + i]   = sp[i];
    target_e[(size_t)b * LL * 3 + i] = st[i];
  }
  for (int p = t; p < LL; p += NT) {
    float x = sp[3*p], y = sp[3*p+1], z = sp[3*p+2];
    npred[(size_t)b * LL + p] = x*x + y*y + z*z;
    x = st[3*p]; y = st[3*p+1]; z = st[3*p+2];
    ntarg[(size_t)b * LL + p] = x*x + y*y + z*z;
  }

  // --- angle & torsion masked sums (mask checks the central vertex only) ---
  float an = 0.f, ad = 0.f, tn = 0.f, td = 0.f;
  for (int j = t; j < LL - 2; j += NT) {
    float w = (j + 1 < cnt) ? 1.f : 0.f;
    float s1, c1, s2, c2;
    angle_sc(sp, j, s1, c1);
    angle_sc(st, j, s2, c2);
    float ds = s1 - s2, dc = c1 - c2;
    an += w * (ds*ds + dc*dc);
    ad += w;
  }
  for (int j = t; j < LL - 3; j += NT) {
    float w = (j + 2 < cnt) ? 1.f : 0.f;
    float s1, c1, s2, c2;
    dihedral_sc(sp, j, s1, c1);
    dihedral_sc(st, j, s2, c2);
    float ds = s1 - s2, dc = c1 - c2;
    tn += w * (ds*ds + dc*dc);
    td += w;
  }

  auto blockReduce = [&](float v) -> float {
    red[t] = v; __syncthreads();
    for (int s = NT / 2; s > 0; s >>= 1) {
      if (t < s) red[t] += red[t + s];
      __syncthreads();
    }
    float r = red[0]; __syncthreads();
    return r;
  };
  float An = blockReduce(an), Ad = blockReduce(ad);
  float Tn = blockReduce(tn), Td = blockReduce(td);

  if (t == 0) {
    float* a = acc + b * 8;
    a[0] = An; a[1] = Ad; a[2] = Tn; a[3] = Td;
    a[4] = 0.f; a[5] = 0.f;          // zero-init atomics for the cdist stage
    cnt_out[b] = cnt;
  }
}

// ---------------------------------------------------------------------------
// Kernel 2: fused pairwise-distance reduction.
// One 16x16 tile per wave; Gram dot-products via V_WMMA_F32_16X16X4_F32.
//   A frag (16x4 f32): lanes 0-15 -> (x,y) of row M=lane (K=0,1)
//                      lanes 16-31 -> (z,0) of row M=lane-16 (K=2,3)
//   B frag (4x16):     symmetric layout over columns.
//   D elem r          -> row i0 + r + 8*(lane>=16), col j0 + (lane&15)
// Tile predicates are wave-uniform, so EXEC is all-1s at the WMMA.
// ---------------------------------------------------------------------------
__global__ void cdist_loss_kernel(const float* __restrict__ pred_e,
                                  const float* __restrict__ target_e,
                                  const float* __restrict__ npred,
                                  const float* __restrict__ ntarg,
                                  const int*   __restrict__ cnt,
                                  float* __restrict__ acc) {
  __shared__ float rd[NT];
  __shared__ float rc[NT];

  const int t    = threadIdx.x;
  const int wave = t >> 5;
  const int lane = t & 31;
  const int b    = blockIdx.x / BLK_PER_BATCH;
  const int tile = (blockIdx.x % BLK_PER_BATCH) * WAVES_PER_BLK + wave;
  const int ti   = tile / TILES, tj = tile % TILES;
  const int i0   = ti * 16, j0 = tj * 16;
  const int c    = cnt[b];

  float dsum = 0.f, csum = 0.f;
  if (i0 < c && j0 < c) {               // wave-uniform: EXEC stays all-1s
    const int hf = lane >> 4;           // 0: K=0..1 half, 1: K=2..3 half
    const int mm = lane & 15;
    const float* Pb = pred_e   + (size_t)b * LL * 3;
    const float* Tb = target_e + (size_t)b * LL * 3;

    const float* pa = Pb + (size_t)(i0 + mm) * 3;   // A rows
    const float* ta = Tb + (size_t)(i0 + mm) * 3;
    const float* pc = Pb + (size_t)(j0 + mm) * 3;   // B columns
    const float* tc = Tb + (size_t)(j0 + mm) * 3;

    v2f a1; a1.x = hf ? pa[2] : pa[0];  a1.y = hf ? 0.f : pa[1];
    v2f a2; a2.x = hf ? ta[2] : ta[0];  a2.y = hf ? 0.f : ta[1];
    v2f b1; b1.x = hf ? pc[2] : pc[0];  b1.y = hf ? 0.f : pc[1];
    v2f b2; b2.x = hf ? tc[2] : tc[0];  b2.y = hf ? 0.f : tc[1];

    v8f g1 = {};
    g1 = __builtin_amdgcn_wmma_f32_16x16x4_f32(false, a1, false, b1,
                                               (short)0, g1, false, false);
    v8f g2 = {};
    g2 = __builtin_amdgcn_wmma_f32_16x16x4_f32(false, a2, false, b2,
                                               (short)0, g2, false, false);

    const float ncol1 = npred[(size_t)b * LL + j0 + mm];
    const float ncol2 = ntarg[(size_t)b * LL + j0 + mm];
    const int   col   = j0 + mm;

#pragma unroll
    for (int r = 0; r < 8; ++r) {
      int   row = i0 + r + 8 * hf;
      float w   = (row < c && col < c) ? 1.f : 0.f;
      float nr1 = npred[(size_t)b * LL + row];
      float nr2 = ntarg[(size_t)b * LL + row];
      float d1  = sqrtf(fmaxf(nr1 + ncol1 - 2.f * g1[r], 1e-12f));
      float d2  = sqrtf(fmaxf(nr2 + ncol2 - 2.f * g2[r], 1e-12f));
      float dd  = d1 - d2;
      dsum += dd * dd * w;
      float cl  = 3.8f - d1;
      csum += ((d1 < 3.8f) && (d1 > 2.0f)) ? cl * cl * w : 0.f;
    }
  }

  rd[t] = dsum; rc[t] = csum;
  __syncthreads();
  for (int s = NT / 2; s > 0; s >>= 1) {
    if (t < s) { rd[t] += rd[t + s]; rc[t] += rc[t + s]; }
    __syncthreads();
  }
  if (t == 0) {
    atomicAdd(&acc[b * 8 + 4], rd[0]);
    atomicAdd(&acc[b * 8 + 5], rc[0]);
  }
}

// ---------------------------------------------------------------------------
// Kernel 3: finalize. denom = cnt^2 + 1e-8 (sum of dmask == (sum m)^2).
// out = [angle, torsion, dist, inter_dist(=dist), clash]
// ---------------------------------------------------------------------------
__global__ void finalize_kernel(const float* __restrict__ acc,
                                const int*   __restrict__ cnt,
                                float* __restrict__ out) {
  __shared__ float sa[BB], stt[BB], sd[BB], sc[BB];
  const int t = threadIdx.x;
  if (t < BB) {
    const float* a = acc + t * 8;
    sa[t]  = a[0] / fmaxf(a[1], 1e-6f);
    stt[t] = a[2] / fmaxf(a[3], 1e-6f);
    float c   = (float)cnt[t];
    float den = c * c + 1e-8f;
    sd[t] = a[4] / den;
    sc[t] = a[5] / den;
  }
  __syncthreads();
  if (t == 0) {
    float A = 0.f, T = 0.f, D = 0.f, C = 0.f;
    for (int i = 0; i < BB; ++i) { A += sa[i]; T += stt[i]; D += sd[i]; C += sc[i]; }
    const float inv = 1.f / (float)BB;
    out[0] = A * inv;
    out[1] = T * inv;
    out[2] = D * inv;
    out[3] = D * inv;   // inter_dist_loss aliases dist_loss in the reference
    out[4] = C * inv;
  }
}

// ---------------------------------------------------------------------------
extern "C" void kernel_launch(void* const* d_in, const int* in_sizes, int n_in,
                              void* d_out, int out_size, void* d_ws, size_t ws_size,
                              hipStream_t stream) {
  const float* pred   = (const float*)d_in[0];   // (16,768,3) f32
  const float* target = (const float*)d_in[1];   // (16,768,3) f32
  const int*   gmask  = (const int*)d_in[2];     // (16,768)   i32
  // d_in[3] (poc) and d_in[4] (poc_mask): reference computes & discards -> skip.

  float* ws       = (float*)d_ws;
  float* pred_e   = ws;                              // B*L*3
  float* target_e = pred_e + (size_t)BB * LL * 3;    // B*L*3
  float* npred    = target_e + (size_t)BB * LL * 3;  // B*L
  float* ntarg    = npred + (size_t)BB * LL;         // B*L
  float* acc      = ntarg + (size_t)BB * LL;         // B*8
  int*   cnt      = (int*)(acc + BB * 8);            // B

  compact_angles_kernel<<<BB, NT, 0, stream>>>(pred, target, gmask,
                                               pred_e, target_e,
                                               npred, ntarg, cnt, acc);
  cdist_loss_kernel<<<BB * BLK_PER_BATCH, NT, 0, stream>>>(pred_e, target_e,
                                                           npred, ntarg, cnt, acc);
  finalize_kernel<<<1, 32, 0, stream>>>(acc, cnt, (float*)d_out);
}